// GAT_32959579029842
// MI455X (gfx1250) — compile-verified
//
#include <hip/hip_runtime.h>
#include <math.h>

// ---------------------------------------------------------------------------
// Problem constants (match reference)
// ---------------------------------------------------------------------------
static constexpr int kNU = 100000;
static constexpr int kNI = 20000;
static constexpr int kE  = 250000;
static constexpr int kH  = 128;     // hidden = HEADS*D = 2*64

typedef float v2f __attribute__((ext_vector_type(2)));
typedef float v8f __attribute__((ext_vector_type(8)));

// ---------------------------------------------------------------------------
// GEMM:  C[M,128] = A[M,K] * B[K,128] + bias   (K in {64,128}, M % 16 == 0)
// Block = 256 threads (8 waves). Block handles a 16-row strip; wave w owns
// the 16x16 tile at columns [16w, 16w+16). Uses V_WMMA_F32_16X16X4_F32.
//
// Fragment layouts per CDNA5 ISA 7.12.2:
//   A (16x4 f32):  lane%16 = M ; VGPR0 = K+{0 | 2 for lanes>=16}, VGPR1 = +1
//   B (4x16 f32):  lane%16 = N ; VGPR0 = row K+{0|2}, VGPR1 = +1
//   C/D (16x16):   VGPR v: lanes 0-15 -> M=v, lanes 16-31 -> M=v+8
//
// LDS stride K+4: keeps float4 stores 16B-aligned (row byte stride 528 for
// K=128) and bank pattern 4*m mod 64 conflict-free across the 16 M-lanes.
// ---------------------------------------------------------------------------
template <int K>
__global__ void __launch_bounds__(256)
gemm_bias_wmma(const float* __restrict__ A, const float* __restrict__ B,
               const float* __restrict__ bias, float* __restrict__ C)
{
    constexpr int STR = K + 4;
    __shared__ float As[16 * STR];
    const int tid     = threadIdx.x;
    const int wave    = tid >> 5;
    const int lane    = tid & 31;
    const int rowBase = blockIdx.x << 4;

    // Stage the 16 x K strip of A into LDS with float4 transfers.
    constexpr int NV4 = 16 * K / 4;         // 256 (K=64) or 512 (K=128)
#pragma unroll
    for (int idx = tid; idx < NV4; idx += 256) {
        const int r  = idx / (K / 4);
        const int c4 = (idx % (K / 4)) << 2;
        const float4 v = *(const float4*)(A + (size_t)(rowBase + r) * K + c4);
        *(float4*)(&As[r * STR + c4]) = v;
    }
    __syncthreads();

    const int m    = lane & 15;
    const int koff = (lane >> 4) << 1;      // 0 for lanes 0-15, 2 for 16-31
    const int col  = (wave << 4) + (lane & 15);

    v8f acc = {0.f, 0.f, 0.f, 0.f, 0.f, 0.f, 0.f, 0.f};
#pragma unroll
    for (int k = 0; k < K; k += 4) {
        v2f a, b;
        a.x = As[m * STR + k + koff];
        a.y = As[m * STR + k + koff + 1];
        b.x = B[(k + koff)     * kH + col];
        b.y = B[(k + koff + 1) * kH + col];
        acc = __builtin_amdgcn_wmma_f32_16x16x4_f32(false, a, false, b,
                                                    (short)0, acc, false, false);
    }

    const float bv     = bias[col];
    const int   rowOff = (lane >> 4) << 3;  // +8 rows for lanes 16-31
#pragma unroll
    for (int v = 0; v < 8; ++v)
        C[(size_t)(rowBase + rowOff + v) * kH + col] = acc[v] + bv;
}

// ---------------------------------------------------------------------------
// Float atomic max via sign-aware integer punning (init with -inf).
// ---------------------------------------------------------------------------
__device__ __forceinline__ void atomicMaxF(float* addr, float val)
{
    if (val >= 0.f) atomicMax((int*)addr, __float_as_int(val));
    else            atomicMin((unsigned int*)addr, __float_as_uint(val));
}

// ---------------------------------------------------------------------------
// Pass 1: per-edge attention logits + segment max.
// One wave per edge; lane handles a float4 (4 dims); head = lane/16.
// logits[e,h] = sum_d att[h,d] * leaky_relu(Tsrc[src,h,d] + Tdst[dst,h,d])
// ---------------------------------------------------------------------------
__global__ void __launch_bounds__(256)
edge_logits_kernel(const float* __restrict__ Tsrc, const float* __restrict__ Tdst,
                   const int* __restrict__ srcIdx, const int* __restrict__ dstIdx,
                   const float* __restrict__ att,   // [2*64]
                   float* __restrict__ logits,      // [E*2]
                   float* __restrict__ rmax,        // [Ndst*2], pre-set to -inf
                   int E)
{
    const int gw   = (int)((blockIdx.x * blockDim.x + threadIdx.x) >> 5);
    const int lane = threadIdx.x & 31;
    if (gw >= E) return;

    const int s = srcIdx[gw];
    const int d = dstIdx[gw];
    const float4 xs = ((const float4*)(Tsrc + (size_t)s * kH))[lane];
    const float4 xd = ((const float4*)(Tdst + (size_t)d * kH))[lane];
    const float4 av = ((const float4*)att)[lane];

    float e, p = 0.f;
    e = xs.x + xd.x; e = (e > 0.f) ? e : 0.2f * e; p += av.x * e;
    e = xs.y + xd.y; e = (e > 0.f) ? e : 0.2f * e; p += av.y * e;
    e = xs.z + xd.z; e = (e > 0.f) ? e : 0.2f * e; p += av.z * e;
    e = xs.w + xd.w; e = (e > 0.f) ? e : 0.2f * e; p += av.w * e;

    // reduce within each half-wave (16 lanes = one head)
    p += __shfl_xor(p, 1);
    p += __shfl_xor(p, 2);
    p += __shfl_xor(p, 4);
    p += __shfl_xor(p, 8);

    if ((lane & 15) == 0) {
        const int h = lane >> 4;
        logits[gw * 2 + h] = p;
        atomicMaxF(&rmax[d * 2 + h], p);
    }
}

// ---------------------------------------------------------------------------
// Pass 2: p = exp(logit - max[dst]);  segment sum of p.  (in-place on logits)
// ---------------------------------------------------------------------------
__global__ void __launch_bounds__(256)
edge_exp_kernel(float* __restrict__ logits, const int* __restrict__ dstIdx,
                const float* __restrict__ rmax, float* __restrict__ rsum, int E)
{
    const int i = blockIdx.x * blockDim.x + threadIdx.x;
    if (i >= E * 2) return;
    const int e = i >> 1, h = i & 1;
    const int d = dstIdx[e];
    const float p = __expf(logits[i] - rmax[d * 2 + h]);
    logits[i] = p;
    atomicAdd(&rsum[d * 2 + h], p);
}

// ---------------------------------------------------------------------------
// Pass 3: out[dst] += alpha * Tsrc[src].  One wave per edge, float4 per lane.
// ---------------------------------------------------------------------------
__global__ void __launch_bounds__(256)
edge_scatter_kernel(const float* __restrict__ Tsrc,
                    const int* __restrict__ srcIdx, const int* __restrict__ dstIdx,
                    const float* __restrict__ pbuf, const float* __restrict__ rsum,
                    float* __restrict__ out, int E)
{
    const int gw   = (int)((blockIdx.x * blockDim.x + threadIdx.x) >> 5);
    const int lane = threadIdx.x & 31;
    if (gw >= E) return;

    const int s = srcIdx[gw];
    const int d = dstIdx[gw];
    const int h = lane >> 4;
    const float alpha = pbuf[gw * 2 + h] / (rsum[d * 2 + h] + 1e-16f);
    const float4 xs = ((const float4*)(Tsrc + (size_t)s * kH))[lane];
    float* o = out + (size_t)d * kH + lane * 4;
    atomicAdd(o + 0, xs.x * alpha);
    atomicAdd(o + 1, xs.y * alpha);
    atomicAdd(o + 2, xs.z * alpha);
    atomicAdd(o + 3, xs.w * alpha);
}

// ---------------------------------------------------------------------------
// Small utility kernels
// ---------------------------------------------------------------------------
__global__ void fill_kernel(float* __restrict__ p, float v, int n)
{
    const int i = blockIdx.x * blockDim.x + threadIdx.x;
    if (i < n) p[i] = v;
}

__global__ void bias_init_kernel(float* __restrict__ out,
                                 const float* __restrict__ bias, int n)
{
    const int i = blockIdx.x * blockDim.x + threadIdx.x;
    if (i < n) out[i] = bias[i & (kH - 1)];
}

__global__ void relu_kernel(float* __restrict__ p, int n)
{
    const int i = blockIdx.x * blockDim.x + threadIdx.x;
    if (i < n) { const float v = p[i]; p[i] = v > 0.f ? v : 0.f; }
}

// ---------------------------------------------------------------------------
// Host orchestration
// ---------------------------------------------------------------------------
extern "C" void kernel_launch(void* const* d_in, const int* in_sizes, int n_in,
                              void* d_out, int out_size, void* d_ws, size_t ws_size,
                              hipStream_t stream)
{
    const float* x_user  = (const float*)d_in[0];
    const float* x_item  = (const float*)d_in[1];
    const float* Wp_user = (const float*)d_in[2];
    const float* bp_user = (const float*)d_in[3];
    const float* Wp_item = (const float*)d_in[4];
    const float* bp_item = (const float*)d_in[5];
    const float* Wl      = (const float*)d_in[6];   // [L,2,128,128]
    const float* bl      = (const float*)d_in[7];   // [L,2,128]
    const float* Wr      = (const float*)d_in[8];
    const float* br      = (const float*)d_in[9];
    const float* att     = (const float*)d_in[10];  // [L,2,2,64]
    const float* biasp   = (const float*)d_in[11];  // [L,2,128]
    const int*   esrc    = (const int*)d_in[12];
    const int*   edst    = (const int*)d_in[13];

    const size_t NUH = (size_t)kNU * kH;
    const size_t NIH = (size_t)kNI * kH;

    float* ws = (float*)d_ws;
    float* zu0    = ws; ws += NUH;
    float* zu1    = ws; ws += NUH;
    float* zi0    = ws; ws += NIH;
    float* zi1    = ws; ws += NIH;
    float* Tsrc_u = ws; ws += NUH;
    float* Tdst_u = ws; ws += NUH;
    float* Tsrc_i = ws; ws += NIH;
    float* Tdst_i = ws; ws += NIH;
    float* pbuf   = ws; ws += (size_t)kE * 2;
    float* rmax_u = ws; ws += (size_t)kNU * 2;
    float* rsum_u = ws; ws += (size_t)kNU * 2;
    float* rmax_i = ws; ws += (size_t)kNI * 2;
    float* rsum_i = ws; ws += (size_t)kNI * 2;

    float* outU = (float*)d_out;          // zu final [NU*128]
    float* outI = (float*)d_out + NUH;    // zi final [NI*128]

    const int edgeWaveGrid = (kE * 32 + 255) / 256;   // one wave per edge
    const int edgeThrGrid  = (kE * 2 + 255) / 256;    // one thread per (edge,head)

    auto gemm = [&](const float* A, const float* B, const float* b, float* C,
                    int M, int K) {
        if (K == 64)
            gemm_bias_wmma<64><<<M / 16, 256, 0, stream>>>(A, B, b, C);
        else
            gemm_bias_wmma<128><<<M / 16, 256, 0, stream>>>(A, B, b, C);
    };
    auto fill = [&](float* p, float v, int n) {
        fill_kernel<<<(n + 255) / 256, 256, 0, stream>>>(p, v, n);
    };

    // ---- Input projections ----
    gemm(x_user, Wp_user, bp_user, zu0, kNU, 64);
    gemm(x_item, Wp_item, bp_item, zi0, kNI, 128);

    float* zu_cur = zu0;
    float* zi_cur = zi0;

    for (int l = 0; l < 2; ++l) {
        const bool last = (l == 1);
        float* zi_next = last ? outI : zi1;
        float* zu_next = last ? outU : zu1;

        // ---- relation 0: users (src) -> items (dst) ----
        {
            const int lr = l * 2 + 0;
            gemm(zu_cur, Wl + (size_t)lr * kH * kH, bl + lr * kH, Tsrc_u, kNU, kH);
            gemm(zi_cur, Wr + (size_t)lr * kH * kH, br + lr * kH, Tdst_i, kNI, kH);
            fill(rmax_i, -INFINITY, kNI * 2);
            fill(rsum_i, 0.f, kNI * 2);
            edge_logits_kernel<<<edgeWaveGrid, 256, 0, stream>>>(
                Tsrc_u, Tdst_i, esrc, edst, att + lr * kH, pbuf, rmax_i, kE);
            edge_exp_kernel<<<edgeThrGrid, 256, 0, stream>>>(
                pbuf, edst, rmax_i, rsum_i, kE);
            bias_init_kernel<<<((int)NIH + 255) / 256, 256, 0, stream>>>(
                zi_next, biasp + lr * kH, (int)NIH);
            edge_scatter_kernel<<<edgeWaveGrid, 256, 0, stream>>>(
                Tsrc_u, esrc, edst, pbuf, rsum_i, zi_next, kE);
            if (!last)
                relu_kernel<<<((int)NIH + 255) / 256, 256, 0, stream>>>(zi_next, (int)NIH);
        }

        // ---- relation 1: items (src) -> users (dst) ----
        {
            const int lr = l * 2 + 1;
            gemm(zi_cur, Wl + (size_t)lr * kH * kH, bl + lr * kH, Tsrc_i, kNI, kH);
            gemm(zu_cur, Wr + (size_t)lr * kH * kH, br + lr * kH, Tdst_u, kNU, kH);
            fill(rmax_u, -INFINITY, kNU * 2);
            fill(rsum_u, 0.f, kNU * 2);
            edge_logits_kernel<<<edgeWaveGrid, 256, 0, stream>>>(
                Tsrc_i, Tdst_u, edst, esrc, att + lr * kH, pbuf, rmax_u, kE);
            edge_exp_kernel<<<edgeThrGrid, 256, 0, stream>>>(
                pbuf, esrc, rmax_u, rsum_u, kE);
            bias_init_kernel<<<((int)NUH + 255) / 256, 256, 0, stream>>>(
                zu_next, biasp + lr * kH, (int)NUH);
            edge_scatter_kernel<<<edgeWaveGrid, 256, 0, stream>>>(
                Tsrc_i, edst, esrc, pbuf, rsum_u, zu_next, kE);
            if (!last)
                relu_kernel<<<((int)NUH + 255) / 256, 256, 0, stream>>>(zu_next, (int)NUH);
        }

        zu_cur = zu_next;
        zi_cur = zi_next;
    }
}